// PETP_Norm_790273982667
// MI455X (gfx1250) — compile-verified
//
#include <hip/hip_runtime.h>
#include <math.h>

// ---------------- problem constants ----------------
#define BATCH   4096
#define FEAT    576            // 64x0e + 64x1o + 64x2e
#define NROWS   (BATCH * 6)    // 24576 rows of (N, FEAT)
#define NCHAN   192            // norm output channels (3 blocks x 64)
#define SROW    194            // padded S row in LDS (bank-conflict free b64 reads)
#define XROW    577            // padded X row in LDS (bank-conflict free per-sample reads)
#define WPAIRS  1179648        // packed-W float2 pairs per variant tensor
#define WFLOATS (2 * WPAIRS)   // floats per packed variant tensor

typedef __attribute__((ext_vector_type(2))) float v2f;
typedef __attribute__((ext_vector_type(8))) float v8f;

// irrep block meta as compile-time constants (folded by the unrolled k loop)
constexpr int DKc[3] = {1, 3, 5};    // irrep dims
constexpr int FBc[3] = {0, 64, 256}; // feature offsets
constexpr int PKc[3] = {0, 1, 4};    // yacc prefix offsets

// ---------------- explicit global-address-space access ----------------
typedef const __attribute__((address_space(1))) float* gcfp;
typedef __attribute__((address_space(1))) float* gfp;
typedef const __attribute__((address_space(1))) v2f* gcv2p;
typedef __attribute__((address_space(1))) v2f* gv2p;

__device__ __forceinline__ float gload(const float* p) { return *(gcfp)p; }
__device__ __forceinline__ void  gstore(float* p, float v) { *(gfp)p = v; }
__device__ __forceinline__ v2f   gload2(const float* p) { return *(gcv2p)p; }
__device__ __forceinline__ void  gstore2(float* p, v2f v) { *(gv2p)p = v; }

// ---------------- variant reconstruction ----------------
// n = ((b*2+i)*3+j) row index.
// v0 self: x   v1 flavor: (Sj-x)/2   v2 nunubar: Si-x   v3 all: (T-Sj-Si+x)/2
__device__ __forceinline__ float variant_val(int v, int n, int f,
                                             const float* __restrict__ x,
                                             const float* __restrict__ Sj,
                                             const float* __restrict__ Si,
                                             const float* __restrict__ T) {
    int b = n / 6;
    int ij = n - b * 6;
    int i = ij / 3;
    int j = ij - i * 3;
    float xs = gload(x + (size_t)n * FEAT + f);
    if (v == 0) return xs;
    if (v == 1) {
        float sj = gload(Sj + ((size_t)b * 2 + i) * FEAT + f);
        return 0.5f * (sj - xs);
    }
    if (v == 2) {
        float si = gload(Si + ((size_t)b * 3 + j) * FEAT + f);
        return si - xs;
    }
    float sj = gload(Sj + ((size_t)b * 2 + i) * FEAT + f);
    float si = gload(Si + ((size_t)b * 3 + j) * FEAT + f);
    float t  = gload(T + (size_t)b * FEAT + f);
    return 0.5f * (t - sj - si + xs);
}

// ---------------- kernel 1: row sums for PE decomposition ----------------
__global__ void petp_sums_kernel(const float* __restrict__ x,
                                 float* __restrict__ Sj,   // (B,2,FEAT)
                                 float* __restrict__ Si,   // (B,3,FEAT)
                                 float* __restrict__ T) {  // (B,FEAT)
    int t = blockIdx.x * blockDim.x + threadIdx.x;
    if (t >= BATCH * FEAT) return;
    int b = t / FEAT;
    int f = t - b * FEAT;
    float v[2][3];
#pragma unroll
    for (int i = 0; i < 2; ++i)
#pragma unroll
        for (int j = 0; j < 3; ++j)
            v[i][j] = gload(x + (((size_t)b * 2 + i) * 3 + j) * FEAT + f);
    float sj0 = v[0][0] + v[0][1] + v[0][2];
    float sj1 = v[1][0] + v[1][1] + v[1][2];
    gstore(Sj + ((size_t)b * 2 + 0) * FEAT + f, sj0);
    gstore(Sj + ((size_t)b * 2 + 1) * FEAT + f, sj1);
#pragma unroll
    for (int j = 0; j < 3; ++j)
        gstore(Si + ((size_t)b * 3 + j) * FEAT + f, v[0][j] + v[1][j]);
    gstore(T + (size_t)b * FEAT + f, sj0 + sj1);
}

// ---------------- kernel 2: pack W into the WMMA A-operand layout ----------------
// W'[k3j][u][half][s16][w] = float2{ W[k][j][u][vv][w], W[k][j][u][vv+1][w] }
// with vv = 4*s16 + 2*half, so each lane's per-K-step A pair is one b64 load.
// Destination pair index == t by construction of the decomposition below.
__global__ void petp_wpack_kernel(const float* __restrict__ W0,
                                  const float* __restrict__ W1,
                                  const float* __restrict__ W2,
                                  const float* __restrict__ W3,
                                  float* __restrict__ Wt) {
    int t = blockIdx.x * blockDim.x + threadIdx.x;
    if (t >= WPAIRS) return;
    int v = blockIdx.y;
    const float* Ws = (v == 0) ? W0 : (v == 1) ? W1 : (v == 2) ? W2 : W3;
    int r   = t;
    int w   = r & 63;  r >>= 6;
    int s16 = r & 15;  r >>= 4;
    int h   = r & 1;   r >>= 1;
    int u   = r & 63;  r >>= 6;
    int k3j = r;                       // 0..8
    int vv  = s16 * 4 + h * 2;
    size_t src = (((size_t)k3j * 64 + u) * 64 + vv) * 64 + w;
    v2f pr;
    pr.x = gload(Ws + src);            // K = vv
    pr.y = gload(Ws + src + 64);       // K = vv+1 (v-stride = 64 floats)
    gstore2(Wt + (size_t)v * WFLOATS + 2 * (size_t)t, pr);
}

// ---------------- kernel 3: fused norm + (S@W) GEMM (WMMA) + apply ----------------
// Grid: NROWS/16 blocks of 128 threads (4 waves). Block owns 16 samples, loops
// over the 4 PE variants accumulating y in registers; wave owns one 16-wide
// w-chunk for all 3 output irreps k.
//
// Per (variant,k,u,wc) tile, V_WMMA_F32_16X16X4_F32 (48 K-steps over jv=0..191):
//   A(16x4) = packed W' tile (one b64 per lane per step),
//   B(4x16) = S[jv, samples] from LDS (ds_load_2addr_b64, 2 steps per op).
//   D layout: VGPR r, lane L -> w = w0 + r + 8*(L>=16), sample = L%16.
// Apply: y[s,w,m] += D[r] * x_v[s,u,m]  (x independent of r -> register reuse).
__global__ void __launch_bounds__(128)
petp_fused_kernel(const float* __restrict__ x,
                  const float* __restrict__ Sj,
                  const float* __restrict__ Si,
                  const float* __restrict__ T,
                  const float* __restrict__ Wt,
                  float* __restrict__ out) {
    __shared__ float S_lds[16 * SROW];   // per-sample norms (192, padded)
    __shared__ float X_lds[16 * XROW];   // variant rows (576, padded)

    const int tid  = threadIdx.x;
    const int lane = tid & 31;
    const int wave = tid >> 5;           // 0..3 -> w-chunk
    const int sL   = lane & 15;
    const int half = lane >> 4;
    const int n0   = blockIdx.x * 16;
    const int w0   = wave * 16;

    float yacc[8][9];
#pragma unroll
    for (int r = 0; r < 8; ++r)
#pragma unroll
        for (int p = 0; p < 9; ++p) yacc[r][p] = 0.0f;

    for (int v = 0; v < 4; ++v) {
        // ---- stage variant rows into LDS (9216/128 = 72 iters, no divergence) ----
        for (int idx = tid; idx < 16 * FEAT; idx += 128) {
            int s = idx / FEAT;
            int f = idx - s * FEAT;
            X_lds[s * XROW + f] = variant_val(v, n0 + s, f, x, Sj, Si, T);
        }
        __syncthreads();
        // ---- per-channel L2 norms from staged rows (3072/128 = 24 iters) ----
        for (int idx = tid; idx < 16 * NCHAN; idx += 128) {
            int s = idx / NCHAN;
            int c = idx - s * NCHAN;
            int d, fbn;
            if (c < 64)       { d = 1; fbn = c; }
            else if (c < 128) { d = 3; fbn = 64 + (c - 64) * 3; }
            else              { d = 5; fbn = 256 + (c - 128) * 5; }
            float acc = 0.0f;
            for (int m = 0; m < d; ++m) {
                float t = X_lds[s * XROW + fbn + m];
                acc += t * t;
            }
            S_lds[s * SROW + c] = sqrtf(acc);
        }
        __syncthreads();

        const float* __restrict__ Wv = Wt + (size_t)v * WFLOATS;
#pragma unroll
        for (int k = 0; k < 3; ++k) {
            const int dk = DKc[k];
            const int fb = FBc[k];
            const int pk = PKc[k];
            for (int u = 0; u < 64; ++u) {
                v8f d = {0.0f, 0.0f, 0.0f, 0.0f, 0.0f, 0.0f, 0.0f, 0.0f};
#pragma unroll
                for (int jj = 0; jj < 3; ++jj) {
                    // packed A base for this lane: [k3j][u][half][s16=0][w0+sL]
                    const float* wp = Wv +
                        2 * (((((size_t)(k * 3 + jj) * 64 + u) * 2 + half) * 16)
                             * 64 + w0 + sL);
                    int soff = sL * SROW + jj * 64 + 2 * half;
#pragma unroll
                    for (int s16 = 0; s16 < 16; ++s16) {
                        v2f a = gload2(wp);          // A pair (K, K+1), one b64
                        v2f b;                       // B = S tile
                        b.x = S_lds[soff];
                        b.y = S_lds[soff + 1];
                        d = __builtin_amdgcn_wmma_f32_16x16x4_f32(
                                false, a, false, b, (short)0, d, false, false);
                        wp += 128;                   // next s16 (stride 64 pairs)
                        soff += 4;
                    }
                }
                // ---- apply: y[s,w,m] += M[w,s] * x_v[s,u,m] ----
                float xv[5];
                const int xbase = sL * XROW + fb + u * dk;
#pragma unroll
                for (int m = 0; m < dk; ++m) xv[m] = X_lds[xbase + m];
#pragma unroll
                for (int r = 0; r < 8; ++r)
#pragma unroll
                    for (int m = 0; m < dk; ++m)
                        yacc[r][pk + m] += d[r] * xv[m];
            }
        }
        __syncthreads();   // done reading LDS for this variant
    }

    // ---- write y (each element exactly once) ----
    const float COEF = 0.009021130259f;   // 1/sqrt(3*64*64)
#pragma unroll
    for (int k = 0; k < 3; ++k) {
        const int dk = DKc[k];
        const int fb = FBc[k];
        const int pk = PKc[k];
#pragma unroll
        for (int r = 0; r < 8; ++r) {
            const int w = w0 + r + 8 * half;
#pragma unroll
            for (int m = 0; m < dk; ++m)
                gstore(out + (size_t)(n0 + sL) * FEAT + fb + w * dk + m,
                       COEF * yacc[r][pk + m]);
        }
    }
}

// ---------------- launcher ----------------
extern "C" void kernel_launch(void* const* d_in, const int* in_sizes, int n_in,
                              void* d_out, int out_size, void* d_ws, size_t ws_size,
                              hipStream_t stream) {
    const float* x  = (const float*)d_in[0];
    const float* W0 = (const float*)d_in[1];   // W_self
    const float* W1 = (const float*)d_in[2];   // W_flavor
    const float* W2 = (const float*)d_in[3];   // W_nunubar
    const float* W3 = (const float*)d_in[4];   // W_all
    float* out = (float*)d_out;

    // workspace: Sj | Si | T (56.6 MB) | packed W' (37.7 MB)  = 94.4 MB
    float* Sjw = (float*)d_ws;
    float* Siw = Sjw + (size_t)BATCH * 2 * FEAT;
    float* Tw  = Siw + (size_t)BATCH * 3 * FEAT;
    float* Wtw = Tw  + (size_t)BATCH * FEAT;

    {
        int n = BATCH * FEAT;
        petp_sums_kernel<<<(n + 255) / 256, 256, 0, stream>>>(x, Sjw, Siw, Tw);
    }
    {
        dim3 grid((WPAIRS + 255) / 256, 4);
        petp_wpack_kernel<<<grid, 256, 0, stream>>>(W0, W1, W2, W3, Wtw);
    }
    petp_fused_kernel<<<NROWS / 16, 128, 0, stream>>>(
        x, Sjw, Siw, Tw, Wtw, out);
}